// Grouper_46875273068857
// MI455X (gfx1250) — compile-verified
//
#include <hip/hip_runtime.h>

typedef float v2f __attribute__((ext_vector_type(2)));
typedef float v8f __attribute__((ext_vector_type(8)));
typedef int v4i __attribute__((ext_vector_type(4)));

#define NPTS  65536
#define NGRP  1024
#define GSIZE 32
#define FEAT  64
#define TKT   128   // threads per top-k block

#if defined(__HIP_DEVICE_COMPILE__) && \
    __has_builtin(__builtin_amdgcn_global_load_async_to_lds_b128) && \
    __has_builtin(__builtin_amdgcn_s_wait_asynccnt)
#define USE_ASYNC_LDS 1
typedef __attribute__((address_space(1))) v4i* gbl_v4i_ptr;  // global (AS1) int4*
typedef __attribute__((address_space(3))) v4i* lds_v4i_ptr;  // LDS    (AS3) int4*
#else
#define USE_ASYNC_LDS 0
#endif

// ---------------- K0: per-point squared norms ----------------
__global__ __launch_bounds__(256) void pn2_kernel(const float* __restrict__ x,
                                                  float* __restrict__ pn2) {
  int n = blockIdx.x * blockDim.x + threadIdx.x;
  if (n < NPTS) {
    float a = x[3 * n], b = x[3 * n + 1], c = x[3 * n + 2];
    pn2[n] = a * a + b * b + c * c;
  }
}

// ---------------- K1: farthest point sampling (single 1024-thread WG) ----------------
// dist[64] lives in VGPRs (fully unrolled); x streams from L2 each iteration.
// Centers written as SoA block: cen[0..1023]=x, [1024..]=y, [2048..]=z, [3072..]=h.
// Tie-break: strict '>' keeps first occurrence (lowest index) like jnp.argmax.
__global__ __launch_bounds__(1024) void fps_kernel(const float* __restrict__ x,
                                                   int* __restrict__ cenIdx,
                                                   float* __restrict__ cenSoA) {
  __shared__ float s_val[1024];
  __shared__ int s_idx[1024];
  __shared__ float s_c[3];
  const int t = threadIdx.x;
  float dist[64];
#pragma unroll
  for (int j = 0; j < 64; ++j) dist[j] = 1e10f;
  int far = 0;
  for (int it = 0; it < NGRP; ++it) {
    if (t == 0) {
      cenIdx[it] = far;
      float cx = x[3 * far], cy = x[3 * far + 1], cz = x[3 * far + 2];
      s_c[0] = cx; s_c[1] = cy; s_c[2] = cz;
      cenSoA[it] = cx;
      cenSoA[NGRP + it] = cy;
      cenSoA[2 * NGRP + it] = cz;
      cenSoA[3 * NGRP + it] = cx * cx + cy * cy + cz * cz;
    }
    __syncthreads();
    const float cx = s_c[0], cy = s_c[1], cz = s_c[2];
    float bv = -1.0f;
    int bi = 0;
#pragma unroll
    for (int j = 0; j < 64; ++j) {
      int n = j * 1024 + t;
      float dx = x[3 * n] - cx;
      float dy = x[3 * n + 1] - cy;
      float dz = x[3 * n + 2] - cz;
      float d = dx * dx + dy * dy;
      d += dz * dz;
      float nd = fminf(dist[j], d);
      dist[j] = nd;
      if (nd > bv) { bv = nd; bi = n; }
    }
    s_val[t] = bv;
    s_idx[t] = bi;
    __syncthreads();
    for (int s = 512; s > 0; s >>= 1) {
      if (t < s) {
        float ov = s_val[t + s];
        int oi = s_idx[t + s];
        if (ov > s_val[t] || (ov == s_val[t] && oi < s_idx[t])) {
          s_val[t] = ov;
          s_idx[t] = oi;
        }
      }
      __syncthreads();
    }
    far = s_idx[0];
  }
}

// ---------------- K2: nearest center per point via V_WMMA_F32_16X16X4_F32 ----------------
// A (16x4): 16 centers x (x,y,z,0). Lanes 0-15 hold K=0,1; lanes 16-31 hold K=2,3.
// B (4x16): 16 points.  VGPR0 = rows K0/K2, VGPR1 = rows K1/K3.
// D (16x16): lane L holds point N=L%16, rows M=i+8*(L>=16) in D[i].
// argmin_g d(p,g) == argmax_g (dot(p,c_g) - |c_g|^2/2).
// The 16KB SoA center table is staged into LDS with GLOBAL_LOAD_ASYNC_TO_LDS_B128
// (ASYNCcnt path) when the builtin is available.
__global__ __launch_bounds__(256) void nearest_kernel(const float* __restrict__ x,
                                                      const float* __restrict__ cenSoA,
                                                      float* __restrict__ nearestOut) {
  __shared__ __align__(16) float s_cen[4 * NGRP];  // x | y | z | h, same layout as cenSoA
  const int t = threadIdx.x;
#if USE_ASYNC_LDS
  // Each lane moves 16B; 256 threads x 4 iters cover 16KB. Layout-preserving copy.
  for (int i = t; i < NGRP; i += 256) {
    __builtin_amdgcn_global_load_async_to_lds_b128(
        (gbl_v4i_ptr)(cenSoA + 4 * i),
        (lds_v4i_ptr)(&s_cen[4 * i]),
        /*offset=*/0, /*cpol=*/0);
  }
  __builtin_amdgcn_s_wait_asynccnt(0);
  __syncthreads();
#else
  for (int i = t; i < 4 * NGRP; i += 256) s_cen[i] = cenSoA[i];
  __syncthreads();
#endif
  const float* s_cx = s_cen;
  const float* s_cy = s_cen + NGRP;
  const float* s_cz = s_cen + 2 * NGRP;
  const float* s_h = s_cen + 3 * NGRP;

  const int wave = t >> 5;
  const int lane = t & 31;
  const int hihalf = lane >> 4;  // 0: lanes 0-15, 1: lanes 16-31
  const int l16 = lane & 15;
  const int tile = blockIdx.x * 8 + wave;
  const int pt = tile * 16 + l16;

  v2f B;
  B.x = x[3 * pt + (hihalf ? 2 : 0)];
  B.y = hihalf ? 0.0f : x[3 * pt + 1];

  float bv = -3.4e38f;
  int bg = 0;
  for (int ct = 0; ct < 64; ++ct) {
    int cb = ct * 16 + l16;
    v2f A;
    A.x = hihalf ? s_cz[cb] : s_cx[cb];
    A.y = hihalf ? 0.0f : s_cy[cb];
    v8f C = {0.f, 0.f, 0.f, 0.f, 0.f, 0.f, 0.f, 0.f};
    v8f D = __builtin_amdgcn_wmma_f32_16x16x4_f32(
        /*neg_a=*/false, A, /*neg_b=*/false, B,
        /*c_mod=*/(short)0, C, /*reuse_a=*/false, /*reuse_b=*/false);
#pragma unroll
    for (int i = 0; i < 8; ++i) {
      int g = ct * 16 + i + hihalf * 8;
      float sc = D[i] - 0.5f * s_h[g];
      if (sc > bv) { bv = sc; bg = g; }  // g monotonic within lane -> first-occurrence tie
    }
  }
  // combine lane L with lane L+16 (other 8 rows), tie -> lower center index
  float ov = __shfl_xor(bv, 16, 32);
  int og = __shfl_xor(bg, 16, 32);
  if (ov > bv || (ov == bv && og < bg)) { bv = ov; bg = og; }
  if (hihalf == 0) nearestOut[pt] = (float)bg;
}

// ---------------- K3: top-32 nearest points per center ----------------
// One block per center. Each thread keeps a sorted ascending 32-list in LDS
// over its stripe (n = j*TKT + t, increasing -> stable ties by index), then a
// 32-round lexicographic-min merge of the 128 sorted heads reproduces
// lax.top_k ordering (ascending distance, lower index first on ties).
__global__ __launch_bounds__(TKT) void topk_kernel(const float* __restrict__ x,
                                                   const float* __restrict__ pn2,
                                                   const float* __restrict__ cenSoA,
                                                   int* __restrict__ nbIdx) {
  __shared__ float s_d[TKT * 32];
  __shared__ int s_i[TKT * 32];
  __shared__ float s_rv[TKT];
  __shared__ int s_ri[TKT];
  __shared__ int s_rt[TKT];
  const int g = blockIdx.x;
  const int t = threadIdx.x;
  const float cx = cenSoA[g];
  const float cy = cenSoA[NGRP + g];
  const float cz = cenSoA[2 * NGRP + g];
  const float cn2 = cenSoA[3 * NGRP + g];
  const int base = t * 32;
  int cnt = 0;
  for (int j = 0; j < NPTS / TKT; ++j) {
    int n = j * TKT + t;
    float px = x[3 * n], py = x[3 * n + 1], pz = x[3 * n + 2];
    float dot = px * cx + py * cy;
    dot += pz * cz;
    float d = -2.0f * dot + cn2;  // same expansion/order as _square_distance
    d += pn2[n];
    int m = cnt;
    if (m == 32) {
      if (!(d < s_d[base + 31])) continue;  // tie at boundary keeps earlier n
      m = 31;
    }
    int p = m;
    while (p > 0 && s_d[base + p - 1] > d) {  // equal d inserts after -> stable
      s_d[base + p] = s_d[base + p - 1];
      s_i[base + p] = s_i[base + p - 1];
      --p;
    }
    s_d[base + p] = d;
    s_i[base + p] = n;
    if (cnt < 32) ++cnt;
  }
  int h = 0;  // head pointer into my sorted list (reads bounded by 31)
  for (int k = 0; k < 32; ++k) {
    s_rv[t] = s_d[base + h];
    s_ri[t] = s_i[base + h];
    s_rt[t] = t;
    __syncthreads();
    for (int s = TKT / 2; s > 0; s >>= 1) {
      if (t < s) {
        float ov = s_rv[t + s];
        int on = s_ri[t + s];
        if (ov < s_rv[t] || (ov == s_rv[t] && on < s_ri[t])) {
          s_rv[t] = ov;
          s_ri[t] = on;
          s_rt[t] = s_rt[t + s];
        }
      }
      __syncthreads();
    }
    if (t == 0) nbIdx[g * 32 + k] = s_ri[0];
    if (t == s_rt[0]) ++h;
    __syncthreads();
  }
}

// ---------------- K4: gather neighbor features (float4, rows are 256B) ----------------
__global__ __launch_bounds__(256) void gather_kernel(const float* __restrict__ features,
                                                     const int* __restrict__ nbIdx,
                                                     float* __restrict__ out) {
  const int g = blockIdx.x;
  const int t = threadIdx.x;
  const float4* f4 = (const float4*)features;
  float4* o4 = (float4*)out;
#pragma unroll
  for (int e = t; e < GSIZE * (FEAT / 4); e += 256) {
    int k = e >> 4, q = e & 15;
    int n = nbIdx[g * GSIZE + k];
    o4[(size_t)(g * GSIZE + k) * 16 + q] = f4[(size_t)n * 16 + q];
  }
}

extern "C" void kernel_launch(void* const* d_in, const int* in_sizes, int n_in,
                              void* d_out, int out_size, void* d_ws, size_t ws_size,
                              hipStream_t stream) {
  const float* x = (const float*)d_in[0];         // [1,65536,3]
  const float* features = (const float*)d_in[1];  // [1,65536,64]
  float* out = (float*)d_out;

  // workspace layout (floats):
  //   cenSoA[4096] (x|y|z|h, 16B aligned) | cenIdx[1024] | pn2[65536] | nbIdx[32768]
  float* ws = (float*)d_ws;
  float* cenSoA = ws;
  int* cenIdx = (int*)(ws + 4 * NGRP);
  float* pn2 = ws + 4 * NGRP + NGRP;
  int* nbIdx = (int*)(ws + 4 * NGRP + NGRP + NPTS);

  float* neighbors = out;                              // 1024*32*64 floats
  float* nearest = out + (size_t)NGRP * GSIZE * FEAT;  // 65536 floats (index values)

  pn2_kernel<<<NPTS / 256, 256, 0, stream>>>(x, pn2);
  fps_kernel<<<1, 1024, 0, stream>>>(x, cenIdx, cenSoA);
  nearest_kernel<<<512, 256, 0, stream>>>(x, cenSoA, nearest);
  topk_kernel<<<NGRP, TKT, 0, stream>>>(x, pn2, cenSoA, nbIdx);
  gather_kernel<<<NGRP, 256, 0, stream>>>(features, nbIdx, neighbors);
}